// PrunedSelfAttentionBlock_14396730376581
// MI455X (gfx1250) — compile-verified
//
#include <hip/hip_runtime.h>
#include <math.h>

// ---------------------------------------------------------------------------
// Problem constants (from reference): B=8, C=512, L=4096, H=8, TOPK=256, M=2048
// ---------------------------------------------------------------------------
constexpr int Bn   = 8;
constexpr int Cn   = 512;
constexpr int Ln   = 4096;
constexpr int Hn   = 8;
constexpr int CHn  = 64;      // C / H
constexpr int TOPK = 256;
constexpr int Mn   = 2048;
constexpr int KP1  = TOPK + 1;   // 257 keys after bias_k/bias_v append

typedef __attribute__((ext_vector_type(2))) float v2f;
typedef __attribute__((ext_vector_type(8))) float v8f;

// ---------------------------------------------------------------------------
// Generic fp32 WMMA GEMM:  Y[b][o][n] = sum_k W[o][k] * X[b][k][n] + bias[o]
// W shared across batch (row-major M x K). Block = 8 waves; each wave computes
// a 32(M) x 64(N) tile: 2 A-fragments x 4 B-fragments -> 8 accumulators,
// K-loop step 4 with v_wmma_f32_16x16x4_f32. Inner loop uses incrementing
// pointers only (no per-iter 64-bit address rebuild). B-fragment loads are
// coalesced across lanes; W and X tiles are L2-resident (192 MB L2).
// Requires M%256==0, N%64==0, K%4==0.
// ---------------------------------------------------------------------------
__global__ __launch_bounds__(256)
void gemm_wmma_f32(const float* __restrict__ W, const float* __restrict__ X,
                   const float* __restrict__ bias, float* __restrict__ Y,
                   int Mdim, int Ndim, int Kdim,
                   long strideX, long strideY, int ldx, int ldy)
{
    const int b    = blockIdx.z;
    const float* Xb = X + (long)b * strideX;
    float*       Yb = Y + (long)b * strideY;

    const int wave = threadIdx.x >> 5;      // 0..7
    const int lane = threadIdx.x & 31;
    const int m0   = blockIdx.y * 256 + wave * 32;   // 32 output rows per wave
    const int n0   = blockIdx.x * 64;

    const int koff = (lane >> 4) << 1;      // A/B: lanes 16-31 carry K+2,K+3
    const int ncol = lane & 15;

    // A-fragment pointers for the two 16-row tiles (8B contiguous per lane)
    const float* wp0 = W + (long)(m0 + (lane & 15)) * Kdim + koff;
    const float* wp1 = wp0 + (long)16 * Kdim;
    // B-fragment pointer (coalesced across lanes 0-15 within each 16-col tile)
    const float* xr  = Xb + (long)koff * ldx + n0 + ncol;
    const long xstep = 4L * (long)ldx;

    v8f a00 = {}; v8f a01 = {}; v8f a02 = {}; v8f a03 = {};
    v8f a10 = {}; v8f a11 = {}; v8f a12 = {}; v8f a13 = {};

    for (int k = 0; k < Kdim; k += 4) {
        v2f fa0; fa0.x = wp0[0]; fa0.y = wp0[1];
        v2f fa1; fa1.x = wp1[0]; fa1.y = wp1[1];

        v2f b0, b1, b2, b3;
        b0.x = xr[0];  b0.y = xr[ldx];
        b1.x = xr[16]; b1.y = xr[ldx + 16];
        b2.x = xr[32]; b2.y = xr[ldx + 32];
        b3.x = xr[48]; b3.y = xr[ldx + 48];

        a00 = __builtin_amdgcn_wmma_f32_16x16x4_f32(false, fa0, false, b0, (short)0, a00, false, false);
        a01 = __builtin_amdgcn_wmma_f32_16x16x4_f32(false, fa0, false, b1, (short)0, a01, false, false);
        a02 = __builtin_amdgcn_wmma_f32_16x16x4_f32(false, fa0, false, b2, (short)0, a02, false, false);
        a03 = __builtin_amdgcn_wmma_f32_16x16x4_f32(false, fa0, false, b3, (short)0, a03, false, false);
        a10 = __builtin_amdgcn_wmma_f32_16x16x4_f32(false, fa1, false, b0, (short)0, a10, false, false);
        a11 = __builtin_amdgcn_wmma_f32_16x16x4_f32(false, fa1, false, b1, (short)0, a11, false, false);
        a12 = __builtin_amdgcn_wmma_f32_16x16x4_f32(false, fa1, false, b2, (short)0, a12, false, false);
        a13 = __builtin_amdgcn_wmma_f32_16x16x4_f32(false, fa1, false, b3, (short)0, a13, false, false);

        wp0 += 4; wp1 += 4; xr += xstep;
    }

    // C/D layout: VGPR r -> (M = tile + r + 8*(lane>=16), N = n-tile + (lane&15))
    const int rb = (lane >> 4) << 3;
#pragma unroll
    for (int r = 0; r < 8; ++r) {
        {
            const int row = m0 + rb + r;
            const float bv = bias[row];
            float* yr = Yb + (long)row * ldy + n0;
            yr[ncol]      = a00[r] + bv;
            yr[16 + ncol] = a01[r] + bv;
            yr[32 + ncol] = a02[r] + bv;
            yr[48 + ncol] = a03[r] + bv;
        }
        {
            const int row = m0 + 16 + rb + r;
            const float bv = bias[row];
            float* yr = Yb + (long)row * ldy + n0;
            yr[ncol]      = a10[r] + bv;
            yr[16 + ncol] = a11[r] + bv;
            yr[32 + ncol] = a12[r] + bv;
            yr[48 + ncol] = a13[r] + bv;
        }
    }
}

// ---------------------------------------------------------------------------
// Channel LayerNorm (over rows, per column), fused optional scale / relu /
// residual. Thread-per-column (fully coalesced, 3 passes; data is L2-resident).
//   y = (x-mu)*rsqrt(var+eps)*g + bp;  y *= (*scale);  relu;  y += resid
// ---------------------------------------------------------------------------
__global__ void chan_ln_fused(const float* __restrict__ Xin, float* __restrict__ Yout,
                              const float* __restrict__ g, const float* __restrict__ bp,
                              const float* __restrict__ scale_ptr,
                              const float* __restrict__ resid,
                              int Crows, long bsX, long bsY, long bsR, int do_relu)
{
    const long t = (long)blockIdx.x * blockDim.x + threadIdx.x;  // over B*L
    const int b = (int)(t / Ln);
    const int l = (int)(t % Ln);
    const float* xc = Xin + (long)b * bsX + l;

    float mu = 0.f;
    for (int r = 0; r < Crows; ++r) mu += xc[(long)r * Ln];
    mu /= (float)Crows;
    float va = 0.f;
    for (int r = 0; r < Crows; ++r) { float d = xc[(long)r * Ln] - mu; va += d * d; }
    va /= (float)Crows;
    const float inv = rsqrtf(va + 1e-5f);
    const float sc  = scale_ptr ? *scale_ptr : 1.0f;

    float* yc = Yout + (long)b * bsY + l;
    const float* rc = resid ? (resid + (long)b * bsR + l) : nullptr;
    for (int r = 0; r < Crows; ++r) {
        float y = (xc[(long)r * Ln] - mu) * inv * g[r] + bp[r];
        y *= sc;
        if (do_relu) y = fmaxf(y, 0.0f);
        if (rc) y += rc[(long)r * Ln];
        yc[(long)r * Ln] = y;
    }
}

// ---------------------------------------------------------------------------
// Per-head L2 normalization in place: normalize X[b, h*64 .. h*64+63, l].
// Thread per (b,h,l); coalesced across l.
// ---------------------------------------------------------------------------
__global__ void l2norm_heads(float* __restrict__ X, long batchStride)
{
    const long t = (long)blockIdx.x * blockDim.x + threadIdx.x;  // B*H*L
    const int l = (int)(t % Ln);
    const long bh = t / Ln;
    const int h = (int)(bh % Hn);
    const int b = (int)(bh / Hn);
    float* base = X + (long)b * batchStride + (long)(h * CHn) * Ln + l;
    float ss = 0.f;
    for (int c = 0; c < CHn; ++c) { float v = base[(long)c * Ln]; ss += v * v; }
    const float inv = 1.0f / fmaxf(sqrtf(ss), 1e-12f);
    for (int c = 0; c < CHn; ++c) base[(long)c * Ln] *= inv;
}

// ---------------------------------------------------------------------------
// q_probe[bh*64+c] = sum_l Qn[b, h*64+c, l]  (Qn contiguous (B,512,L) rows).
// One wave per row; wave32 shuffle reduction.
// ---------------------------------------------------------------------------
__global__ void qprobe_kernel(const float* __restrict__ Qn, float* __restrict__ qprobe)
{
    const int wid  = (int)((blockIdx.x * blockDim.x + threadIdx.x) >> 5);  // B*C rows
    const int lane = threadIdx.x & 31;
    const float* rowp = Qn + (long)wid * Ln;
    float s = 0.f;
    for (int l = lane; l < Ln; l += 32) s += rowp[l];
    for (int off = 16; off; off >>= 1) s += __shfl_xor(s, off, 32);
    if (lane == 0) qprobe[wid] = s;
}

// ---------------------------------------------------------------------------
// score[bh, l] = sum_c qprobe[bh,c] * (|kn| + kn)   (kn = normalized K)
// ---------------------------------------------------------------------------
__global__ void score_kernel(const float* __restrict__ Kn, const float* __restrict__ qprobe,
                             float* __restrict__ score)
{
    const int bh = blockIdx.y;
    const int b = bh >> 3, h = bh & 7;
    __shared__ float qp[CHn];
    if (threadIdx.x < CHn) qp[threadIdx.x] = qprobe[bh * CHn + threadIdx.x];
    __syncthreads();
    const int l = blockIdx.x * 256 + threadIdx.x;
    const float* base = Kn + (long)b * (2 * Cn) * Ln + (long)(h * CHn) * Ln + l;
    float s = 0.f;
    for (int c = 0; c < CHn; ++c) { float v = base[(long)c * Ln]; s += qp[c] * (fabsf(v) + v); }
    score[(long)bh * Ln + l] = s;
}

// ---------------------------------------------------------------------------
// Deterministic top-256 set-select per row via 4x8-bit radix refinement, then
// in-order wave-ballot compaction (order fixed -> bit-deterministic output).
// ---------------------------------------------------------------------------
__device__ __forceinline__ unsigned f2key(float x)
{
    unsigned u = __float_as_uint(x);
    return (u & 0x80000000u) ? ~u : (u | 0x80000000u);
}

__global__ void topk_kernel(const float* __restrict__ score, int* __restrict__ idx)
{
    const int row = blockIdx.x;           // B*H rows
    const int tid = threadIdx.x;          // 256 threads
    const float* s = score + (long)row * Ln;

    __shared__ unsigned hist[256];
    __shared__ unsigned sh_pref;
    __shared__ int sh_k;

    unsigned pref = 0, prefMask = 0;
    int kneed = TOPK;

    for (int shift = 24; shift >= 0; shift -= 8) {
        hist[tid] = 0;
        __syncthreads();
        for (int i = tid; i < Ln; i += 256) {
            unsigned u = f2key(s[i]);
            if ((u & prefMask) == pref) atomicAdd(&hist[(u >> shift) & 255], 1u);
        }
        __syncthreads();
        if (tid == 0) {
            int cum = 0, d = 255;
            for (; d > 0; --d) { cum += (int)hist[d]; if (cum >= kneed) break; }
            if (cum < kneed) cum += (int)hist[0];           // d == 0 fallthrough
            sh_pref = pref | ((unsigned)d << shift);
            sh_k    = kneed - (cum - (int)hist[d]);
        }
        __syncthreads();
        pref = sh_pref;
        kneed = sh_k;
        prefMask |= (255u << shift);
        __syncthreads();
    }

    // pref = exact threshold key; (TOPK - kneed) strictly-greater, kneed ties.
    const int wave = tid >> 5, lane = tid & 31;
    if (wave == 0) {
        int gtBase = 0, eqBase = 0;
        const int basEq = TOPK - kneed;
        for (int i0 = 0; i0 < Ln; i0 += 32) {
            const int i = i0 + lane;
            const unsigned u = f2key(s[i]);
            const unsigned long long mGT = __ballot(u > pref);
            const unsigned long long mEQ = __ballot(u == pref);
            const unsigned long long lt = ((unsigned long long)1 << lane) - 1;
            if (u > pref) {
                idx[row * TOPK + gtBase + __popcll(mGT & lt)] = i;
            } else if (u == pref) {
                const int p = eqBase + __popcll(mEQ & lt);
                if (p < kneed) idx[row * TOPK + basEq + p] = i;
            }
            gtBase += __popcll(mGT);
            eqBase += __popcll(mEQ);
        }
    }
}

// ---------------------------------------------------------------------------
// Gather selected key/value columns:
//   KnG[b,ch,j] = Kn[b,ch,idx[bh,j]] ; VG[b,ch,j] = Vln[b,ch,idx[bh,j]]
// ---------------------------------------------------------------------------
__global__ void gather_kernel(const float* __restrict__ KV, const int* __restrict__ idx,
                              float* __restrict__ KnG, float* __restrict__ VG)
{
    const long t = (long)blockIdx.x * blockDim.x + threadIdx.x;   // B*C*TOPK
    const int j  = (int)(t % TOPK);
    const long bc = t / TOPK;
    const int ch = (int)(bc % Cn);
    const int b  = (int)(bc / Cn);
    const int h  = ch >> 6;
    const int l  = idx[(b * Hn + h) * TOPK + j];
    KnG[((long)b * Cn + ch) * TOPK + j] = KV[((long)b * 2 * Cn + ch) * Ln + l];
    VG [((long)b * Cn + ch) * TOPK + j] = KV[((long)b * 2 * Cn + Cn + ch) * Ln + l];
}

// ---------------------------------------------------------------------------
// Append bias_k / bias_v as key column 256.
// ---------------------------------------------------------------------------
__global__ void append_bias(const float* __restrict__ bk, const float* __restrict__ bv,
                            float* __restrict__ kpT, float* __restrict__ vpT)
{
    const int t = blockIdx.x * blockDim.x + threadIdx.x;   // B*C
    const int o = t % Cn, b = t / Cn;
    kpT[((long)b * Cn + o) * KP1 + TOPK] = bk[o];
    vpT[((long)b * Cn + o) * KP1 + TOPK] = bv[o];
}

// ---------------------------------------------------------------------------
// Attention over 257 keys. Block = 8 waves, one query per wave.
// logits -> max -> exp -> sum -> o[c] = (1/sum) * sum_k e_k * v[k,c]
// ---------------------------------------------------------------------------
__global__ __launch_bounds__(256)
void attention_kernel(const float* __restrict__ QP, const float* __restrict__ kpT,
                      const float* __restrict__ vpT, float* __restrict__ Ocl)
{
    const int b = blockIdx.z, h = blockIdx.y;
    const int wave = threadIdx.x >> 5, lane = threadIdx.x & 31;
    const int l0 = blockIdx.x * 8;
    const int l  = l0 + wave;

    __shared__ float qs[8][CHn];
    __shared__ float at[8][KP1 + 7];
    __shared__ float os[8][CHn];

    const float* qb = QP + ((long)b * Cn + h * CHn) * Ln + l;
    qs[wave][lane]      = qb[(long)lane * Ln];
    qs[wave][lane + 32] = qb[(long)(lane + 32) * Ln];
    __syncthreads();

    const float* kB = kpT + ((long)b * Cn + h * CHn) * KP1;
    const float* vB = vpT + ((long)b * Cn + h * CHn) * KP1;

    float mmax = -1e30f;
    for (int k = lane; k < KP1; k += 32) {
        float d = 0.f;
#pragma unroll 8
        for (int c = 0; c < CHn; ++c) d += qs[wave][c] * kB[(long)c * KP1 + k];
        d *= 0.125f;                         // 1/sqrt(CH)
        at[wave][k] = d;
        mmax = fmaxf(mmax, d);
    }
    for (int off = 16; off; off >>= 1) mmax = fmaxf(mmax, __shfl_xor(mmax, off, 32));

    float ssum = 0.f;
    for (int k = lane; k < KP1; k += 32) {
        float e = __expf(at[wave][k] - mmax);
        at[wave][k] = e;
        ssum += e;
    }
    for (int off = 16; off; off >>= 1) ssum += __shfl_xor(ssum, off, 32);
    const float inv = 1.0f / ssum;

    float a0 = 0.f, a1 = 0.f;
    const float* v0 = vB + (long)lane * KP1;
    const float* v1 = vB + (long)(lane + 32) * KP1;
    for (int k = 0; k < KP1; ++k) {
        const float w = at[wave][k];
        a0 += w * v0[k];
        a1 += w * v1[k];
    }
    os[wave][lane]      = a0 * inv;
    os[wave][lane + 32] = a1 * inv;
    __syncthreads();

    // coalesced write: 8 consecutive l per channel written by 8 adjacent threads
    float* ob = Ocl + ((long)b * Cn + h * CHn) * Ln + l0;
    for (int e = threadIdx.x; e < CHn * 8; e += 256) {
        const int c = e >> 3, q = e & 7;
        ob[(long)c * Ln + q] = os[q][c];
    }
}

// ---------------------------------------------------------------------------
// Host-side orchestration
// ---------------------------------------------------------------------------
extern "C" void kernel_launch(void* const* d_in, const int* in_sizes, int n_in,
                              void* d_out, int out_size, void* d_ws, size_t ws_size,
                              hipStream_t stream)
{
    (void)in_sizes; (void)n_in; (void)out_size; (void)ws_size;

    const float* x         = (const float*)d_in[0];
    const float* q_conv_w  = (const float*)d_in[1];
    const float* q_conv_b  = (const float*)d_in[2];
    const float* q_ln_g    = (const float*)d_in[3];
    const float* q_ln_b    = (const float*)d_in[4];
    const float* kv_conv_w = (const float*)d_in[5];
    const float* kv_conv_b = (const float*)d_in[6];
    const float* k_ln_g    = (const float*)d_in[7];
    const float* k_ln_b    = (const float*)d_in[8];
    const float* v_ln_g    = (const float*)d_in[9];
    const float* v_ln_b    = (const float*)d_in[10];
    const float* in_proj_w = (const float*)d_in[11];
    const float* in_proj_b = (const float*)d_in[12];
    const float* bias_k    = (const float*)d_in[13];
    const float* bias_v    = (const float*)d_in[14];
    const float* out_proj_w= (const float*)d_in[15];
    const float* out_proj_b= (const float*)d_in[16];
    const float* attn_ln_g = (const float*)d_in[17];
    const float* attn_ln_b = (const float*)d_in[18];
    const float* gamma     = (const float*)d_in[19];
    const float* w1        = (const float*)d_in[20];
    const float* b1        = (const float*)d_in[21];
    const float* n1_g      = (const float*)d_in[22];
    const float* n1_b      = (const float*)d_in[23];
    const float* w2        = (const float*)d_in[24];
    const float* b2        = (const float*)d_in[25];
    const float* n2_g      = (const float*)d_in[26];
    const float* n2_b      = (const float*)d_in[27];

    float* out = (float*)d_out;

    // ---- workspace carve-up (floats) ----
    float* wsf = (float*)d_ws;
    size_t off = 0;
    float* Qn   = wsf + off; off += (size_t)Bn * Cn * Ln;       // Q -> normalized Q; reused as Ocl
    float* KV   = wsf + off; off += (size_t)Bn * 2 * Cn * Ln;   // K rows then V rows per batch
    float* QP   = wsf + off; off += (size_t)Bn * Cn * Ln;       // qp^T; reused as H2pre
    float* H1   = wsf + off; off += (size_t)Bn * Mn * Ln;       // FFN hidden
    float* A    = wsf + off; off += (size_t)Bn * Cn * Ln;       // aPre -> a (in place)
    float* KnG  = wsf + off; off += (size_t)Bn * Cn * TOPK;
    float* VG   = wsf + off; off += (size_t)Bn * Cn * TOPK;
    float* kpT  = wsf + off; off += (size_t)Bn * Cn * KP1;
    float* vpT  = wsf + off; off += (size_t)Bn * Cn * KP1;
    float* scoreB  = wsf + off; off += (size_t)Bn * Hn * Ln;
    float* qprobeB = wsf + off; off += (size_t)Bn * Hn * CHn;
    int*   idxB    = (int*)(wsf + off); off += (size_t)Bn * Hn * TOPK;

    const long sCL  = (long)Cn * Ln;
    const long s2CL = (long)2 * Cn * Ln;
    const long sML  = (long)Mn * Ln;

    // 1) Q = Wq @ x + bq ; KV = Wkv @ x + bkv
    gemm_wmma_f32<<<dim3(Ln / 64, Cn / 256, Bn), 256, 0, stream>>>(
        q_conv_w, x, q_conv_b, Qn, Cn, Ln, Cn, sCL, sCL, Ln, Ln);
    gemm_wmma_f32<<<dim3(Ln / 64, (2 * Cn) / 256, Bn), 256, 0, stream>>>(
        kv_conv_w, x, kv_conv_b, KV, 2 * Cn, Ln, Cn, sCL, s2CL, Ln, Ln);

    // 2) channel LayerNorms (Q in place; K = first half of KV; V = second half)
    const int lnBlocks = (Bn * Ln) / 256;
    chan_ln_fused<<<lnBlocks, 256, 0, stream>>>(Qn, Qn, q_ln_g, q_ln_b, nullptr, nullptr,
                                                Cn, sCL, sCL, 0, 0);
    chan_ln_fused<<<lnBlocks, 256, 0, stream>>>(KV, KV, k_ln_g, k_ln_b, nullptr, nullptr,
                                                Cn, s2CL, s2CL, 0, 0);
    chan_ln_fused<<<lnBlocks, 256, 0, stream>>>(KV + sCL, KV + sCL, v_ln_g, v_ln_b, nullptr, nullptr,
                                                Cn, s2CL, s2CL, 0, 0);

    // 3) per-head L2 normalization of Q and K (in place)
    const int l2Blocks = (Bn * Hn * Ln) / 256;
    l2norm_heads<<<l2Blocks, 256, 0, stream>>>(Qn, sCL);
    l2norm_heads<<<l2Blocks, 256, 0, stream>>>(KV, s2CL);   // K half only (rows 0..511)

    // 4) probe + score + top-k + gather
    qprobe_kernel<<<(Bn * Cn * 32) / 256, 256, 0, stream>>>(Qn, qprobeB);
    score_kernel<<<dim3(Ln / 256, Bn * Hn), 256, 0, stream>>>(KV, qprobeB, scoreB);
    topk_kernel<<<Bn * Hn, 256, 0, stream>>>(scoreB, idxB);
    gather_kernel<<<(Bn * Cn * TOPK) / 256, 256, 0, stream>>>(KV, idxB, KnG, VG);

    // 5) in-projections (qp over all 4096 columns; kp/vp over 256 gathered columns)
    const float* wq = in_proj_w;                   const float* bq  = in_proj_b;
    const float* wk = in_proj_w + (size_t)Cn*Cn;   const float* bkk = in_proj_b + Cn;
    const float* wv = in_proj_w + (size_t)2*Cn*Cn; const float* bvv = in_proj_b + 2*Cn;

    gemm_wmma_f32<<<dim3(Ln / 64, Cn / 256, Bn), 256, 0, stream>>>(
        wq, Qn, bq, QP, Cn, Ln, Cn, sCL, sCL, Ln, Ln);
    gemm_wmma_f32<<<dim3(TOPK / 64, Cn / 256, Bn), 256, 0, stream>>>(
        wk, KnG, bkk, kpT, Cn, TOPK, Cn, (long)Cn * TOPK, (long)Cn * KP1, TOPK, KP1);
    gemm_wmma_f32<<<dim3(TOPK / 64, Cn / 256, Bn), 256, 0, stream>>>(
        wv, VG, bvv, vpT, Cn, TOPK, Cn, (long)Cn * TOPK, (long)Cn * KP1, TOPK, KP1);
    append_bias<<<(Bn * Cn) / 256, 256, 0, stream>>>(bias_k, bias_v, kpT, vpT);

    // 6) softmax attention -> Ocl (reuses Qn buffer)
    float* Ocl = Qn;
    attention_kernel<<<dim3(Ln / 8, Hn, Bn), 256, 0, stream>>>(QP, kpT, vpT, Ocl);

    // 7) out-projection, then a = LN(aPre)*gamma + x (in place in A)
    gemm_wmma_f32<<<dim3(Ln / 64, Cn / 256, Bn), 256, 0, stream>>>(
        out_proj_w, Ocl, out_proj_b, A, Cn, Ln, Cn, sCL, sCL, Ln, Ln);
    chan_ln_fused<<<lnBlocks, 256, 0, stream>>>(A, A, attn_ln_g, attn_ln_b, gamma, x,
                                                Cn, sCL, sCL, sCL, 0);

    // 8) FFN: h1 = relu(LN(W1 @ a + b1)) ; h2 = relu(LN(W2 @ h1 + b2)) + a
    gemm_wmma_f32<<<dim3(Ln / 64, Mn / 256, Bn), 256, 0, stream>>>(
        w1, A, b1, H1, Mn, Ln, Cn, sCL, sML, Ln, Ln);
    chan_ln_fused<<<lnBlocks, 256, 0, stream>>>(H1, H1, n1_g, n1_b, nullptr, nullptr,
                                                Mn, sML, sML, 0, 1);
    float* H2 = QP;  // reuse
    gemm_wmma_f32<<<dim3(Ln / 64, Cn / 256, Bn), 256, 0, stream>>>(
        w2, H1, b2, H2, Cn, Ln, Mn, sML, sCL, Ln, Ln);
    chan_ln_fused<<<lnBlocks, 256, 0, stream>>>(H2, out, n2_g, n2_b, nullptr, A,
                                                Cn, sCL, sCL, sCL, 1);
}